// Interpolation_83133386981479
// MI455X (gfx1250) — compile-verified
//
#include <hip/hip_runtime.h>

// Problem constants (from the reference)
#define SH     4      // shells
#define VS     90     // dirs per shell (K)
#define VG     642    // grid vertices (M)
#define FIN    8      // feature channels
#define NBATCH 2      // batch
#define XYZ    1728   // 12*12*12 spatial (N)

#define KSTEPS 23     // ceil(90/4) K-steps of V_WMMA_F32_16X16X4_F32
#define PQ     11     // ceil(642/64) M-quad-tiles (64 rows each, padded to 704)

// A-fragment buffer in d_ws: [SH][PQ][KSTEPS][32 lanes][8 floats]
// lane's 8 floats = { (mt=0..3) x (k-parity 0..1) }, k = kstep*4 + 2*(lane>=16) + parity
#define AFRAG_FLOATS (SH * PQ * KSTEPS * 32 * 8)   // 259,072 floats = 1,036,288 B

typedef float v2f __attribute__((ext_vector_type(2)));
typedef float v4f __attribute__((ext_vector_type(4)));
typedef float v8f __attribute__((ext_vector_type(8)));

// Pre-pass: scatter W[S,90,642] into per-lane WMMA A-fragments (zero-padded for
// k>=90 / p>=642). Turns 8 scalar A-loads per K-step into 2 coalesced b128 loads.
__global__ __launch_bounds__(256) void build_afrag_kernel(const float* __restrict__ W,
                                                          float* __restrict__ A) {
    const int i = blockIdx.x * 256 + threadIdx.x;
    if (i >= AFRAG_FLOATS) return;
    const int f8    = i & 7;                 // {m-tile, k-parity}
    const int lane  = (i >> 3) & 31;
    const int rest  = i >> 8;
    const int kstep = rest % KSTEPS;
    const int pq    = (rest / KSTEPS) % PQ;
    const int s     = rest / (KSTEPS * PQ);
    const int mt    = f8 >> 1;
    const int kpar  = f8 & 1;
    const int k     = kstep * 4 + 2 * (lane >> 4) + kpar;
    const int p     = pq * 64 + mt * 16 + (lane & 15);
    A[i] = (k < VS && p < VG) ? W[(size_t)(s * VS + k) * VG + p] : 0.0f;
}

// One wave computes a 64(M=p) x 32(N=xyz) tile of one (b,f,s) GEMM:
//   Y[p, n] = sum_k W[s, k, p] * X[b,f,s, k, n]
// 4 M-tiles x 2 N-tiles; per K-step: 2 b128 A-loads + 4 b32 B-loads -> 8 WMMAs.
__global__ __launch_bounds__(128) void interp_wmma_kernel(
    const float* __restrict__ x,      // [NBATCH, FIN, SH, VS, XYZ]
    const float* __restrict__ Afrag,  // packed A fragments (d_ws)
    float* __restrict__ out)          // [NBATCH, SH, FIN, VG, XYZ]
{
    const int lane   = threadIdx.x & 31;
    const int wave   = threadIdx.x >> 5;
    const int nChunk = blockIdx.x * 4 + wave;        // 32 N-columns per wave
    if (nChunk >= XYZ / 32) return;                  // wave-uniform exit, EXEC all-1s inside

    const int pq = blockIdx.y;                       // 11 quad-tiles of 64 M-rows
    const int z  = blockIdx.z;                       // 64 = NBATCH*FIN*SH
    const int b  = z >> 5;
    const int f  = (z >> 2) & 7;
    const int s  = z & 3;

    const float* __restrict__ xBase = x   + (size_t)((b * FIN + f) * SH + s) * (VS * XYZ);
    float*       __restrict__ oBase = out + (size_t)((b * SH + s) * FIN + f) * ((size_t)VG * XYZ);
    const float* __restrict__ aBase = Afrag + ((size_t)(s * PQ + pq) * KSTEPS * 32 + lane) * 8;

    const int l16   = lane & 15;                     // M index (A) / N index (B, C/D)
    const int lhalf = lane >> 4;                     // K half (A/B), M half (C/D)
    const int nBase = nChunk * 32;

    const float* __restrict__ xp = xBase + (size_t)(2 * lhalf) * XYZ + nBase + l16;
    const float* __restrict__ xq = xBase + nBase + l16;   // tail (clamped k)

    v8f acc[4][2];
    #pragma unroll
    for (int m = 0; m < 4; ++m)
        #pragma unroll
        for (int t = 0; t < 2; ++t)
            acc[m][t] = (v8f){0.f,0.f,0.f,0.f,0.f,0.f,0.f,0.f};

    #pragma unroll
    for (int ks = 0; ks < KSTEPS; ++ks) {
        // A fragments: one contiguous 32B per lane -> two b128 loads, immediate offsets.
        const v4f alo = *(const v4f*)(aBase + (size_t)ks * 256);
        const v4f ahi = *(const v4f*)(aBase + (size_t)ks * 256 + 4);
        v2f a[4];
        a[0] = (v2f){alo.x, alo.y};
        a[1] = (v2f){alo.z, alo.w};
        a[2] = (v2f){ahi.x, ahi.y};
        a[3] = (v2f){ahi.z, ahi.w};

        // B fragments: lanes contiguous in n (coalesced). Tail step (k=88..91):
        // upper-half lanes have A=0 from fragment padding, so clamp their k to 89
        // (in-bounds address, value annihilated). Branch resolved at compile time.
        v2f bv[2];
        if (ks < KSTEPS - 1) {
            const int k0 = ks * 4;
            #pragma unroll
            for (int t = 0; t < 2; ++t) {
                bv[t].x = xp[(size_t)k0 * XYZ + t * 16];
                bv[t].y = xp[(size_t)(k0 + 1) * XYZ + t * 16];
            }
        } else {
            const int kc0 = 88 + lhalf;              // 88 / 89 (clamped from 90)
            const int kc1 = 89;                      // 89 / 89 (clamped from 91)
            #pragma unroll
            for (int t = 0; t < 2; ++t) {
                bv[t].x = xq[(size_t)kc0 * XYZ + t * 16];
                bv[t].y = xq[(size_t)kc1 * XYZ + t * 16];
            }
        }

        #pragma unroll
        for (int t = 0; t < 2; ++t)
            #pragma unroll
            for (int m = 0; m < 4; ++m)
                acc[m][t] = __builtin_amdgcn_wmma_f32_16x16x4_f32(false, a[m], false, bv[t],
                                                                  (short)0, acc[m][t], false, false);
    }

    // Epilogue. Rows pp = pq*64 + m*16 + r + 8*lhalf, cols nBase + t*16 + l16.
    // Output streamed once (284 MB) -> non-temporal so x stays resident in L2.
    if (pq != PQ - 1) {
        // Wave-uniform fast path: all 64 rows valid, zero per-lane guards.
        #pragma unroll
        for (int m = 0; m < 4; ++m) {
            #pragma unroll
            for (int r = 0; r < 8; ++r) {
                const int pp = pq * 64 + m * 16 + r + 8 * lhalf;
                float* row = oBase + (size_t)pp * XYZ + nBase + l16;
                __builtin_nontemporal_store(acc[m][0][r], row);
                __builtin_nontemporal_store(acc[m][1][r], row + 16);
            }
        }
    } else {
        // Last quad-tile: only rows 640, 641 exist -> m=0, lhalf=0, r<2.
        if (lane < 16) {
            #pragma unroll
            for (int r = 0; r < 2; ++r) {
                float* row = oBase + (size_t)(640 + r) * XYZ + nBase + l16;
                __builtin_nontemporal_store(acc[0][0][r], row);
                __builtin_nontemporal_store(acc[0][1][r], row + 16);
            }
        }
    }
}

extern "C" void kernel_launch(void* const* d_in, const int* in_sizes, int n_in,
                              void* d_out, int out_size, void* d_ws, size_t ws_size,
                              hipStream_t stream) {
    const float* x = (const float*)d_in[0];   // [2, 8, 360, 12,12,12] fp32
    const float* W = (const float*)d_in[1];   // [4, 90, 642] fp32
    // d_in[2] = shell_inverse: sorted equal-count repeats -> static, unused.
    float* out = (float*)d_out;               // [2, 32, 642, 12,12,12] fp32

    float* Afrag = (float*)d_ws;              // needs 1,036,288 B

    // Pre-pass (re-run every call; deterministic, overwrites ws).
    build_afrag_kernel<<<(AFRAG_FLOATS + 255) / 256, 256, 0, stream>>>(W, Afrag);

    // Main GEMM: grid.x = ceil(54 nChunks / 4 waves) = 14,
    //            grid.y = 11 M-quad-tiles, grid.z = 64 independent GEMMs.
    dim3 grid(14, 11, 64);
    dim3 block(128);
    interp_wmma_kernel<<<grid, block, 0, stream>>>(x, Afrag, out);
}